// ParallelAttention_84885733638391
// MI455X (gfx1250) — compile-verified
//
#include <hip/hip_runtime.h>
#include <hip/hip_bf16.h>
#include <cstdint>

#define SEQ   2048
#define BATCH 2
#define HID   2048
#define HEADS 16
#define HD    128
#define MTOT  (SEQ * BATCH)     // 4096
#define NQKV  (3 * HID)         // 6144
#define BHSD  ((size_t)BATCH * HEADS * SEQ * HD)  // 8388608 elems per Q/K/V buf

#define KC      16              // K-chunk staged per TDM transfer
#define KCHUNKS (HID / KC)      // 128
#define LDSTR   20              // padded LDS row stride in floats (16 data + 4 pad)

typedef __attribute__((ext_vector_type(2))) float v2f;
typedef __attribute__((ext_vector_type(8))) float v8f;
typedef __attribute__((ext_vector_type(4))) unsigned int u32x4;
typedef __attribute__((ext_vector_type(8))) int i32x8;
typedef __attribute__((ext_vector_type(4))) int i32x4;

#if defined(__has_builtin)
#if __has_builtin(__builtin_amdgcn_tensor_load_to_lds) && \
    __has_builtin(__builtin_amdgcn_s_wait_tensorcnt)
#define USE_TDM 1
#endif
#endif

// LDS aperture: generic pointer's low 32 bits are the wave-relative LDS byte offset.
static __device__ inline unsigned lds_off_of(const void* p) {
    return (unsigned)(uintptr_t)p;
}

#ifdef USE_TDM
// One TDM transfer of a 2D fp32 tile: rows x 16 elems, row stride 2048 elems in
// memory, written to LDS with 4-DWORD padding after every 16 DWORDs (stride 20).
// D# layout per CDNA5 ISA sec 8.3/8.4.
static __device__ inline void tdm_load_tile(unsigned lds_byte_off,
                                            const float* gptr,
                                            int tile_rows, int tensor_rows) {
    const unsigned long long ga = (unsigned long long)(uintptr_t)gptr;
    u32x4 g0;
    g0[0] = 1u;                                        // count=1, user descriptor
    g0[1] = lds_byte_off;                              // lds_addr
    g0[2] = (unsigned)(ga & 0xffffffffu);              // global_addr[31:0]
    g0[3] = (unsigned)((ga >> 32) & 0x01ffffffu)       // global_addr[56:32]
            | (2u << 30);                              // type=2 ("image")
    i32x8 g1;
    g1[0] = (2 << 16)     // data_size = 4B
          | (1 << 20)     // pad_enable
          | (3 << 22)     // pad_interval: 16 DWORDs
          | (3 << 25);    // pad_amount:   4 DWORDs
    g1[1] = (int)((HID & 0xffffu) << 16);              // tensor_dim0[15:0]
    g1[2] = (int)((HID >> 16) |
                  (((unsigned)tensor_rows & 0xffffu) << 16));  // dim0 hi | dim1 lo
    g1[3] = (int)(((unsigned)tensor_rows >> 16) | (16u << 16));// dim1 hi | tile_dim0=16
    g1[4] = tile_rows;                                 // tile_dim1 (tile_dim2=0)
    g1[5] = HID;                                       // tensor_dim0_stride[31:0]
    g1[6] = 0;
    g1[7] = 0;
    const i32x4 z4 = {0, 0, 0, 0};
    const i32x8 z8 = {0, 0, 0, 0, 0, 0, 0, 0};
    // amdgpu-toolchain (clang-23) 6-arg form: (g0, g1, g2, g3, g4, cpol)
    __builtin_amdgcn_tensor_load_to_lds(g0, g1, z4, z4, z8, 0);
}
#endif

// -------------------- Staged fp32 WMMA GEMM core --------------------
// Block: 256 threads = 8 waves, tile M=128 x N=64. Wave w owns rows
// [w*16, w*16+16). A/B staged in LDS (TDM double-buffered), fp32 WMMA 16x16x4.
#define GEMM_MAINLOOP(X, W, mBlock, n0, nrows_of_W, acc)                        \
    __shared__ float ldsA[2][128 * LDSTR];                                      \
    __shared__ float ldsB[2][64 * LDSTR];                                       \
    const int wave = threadIdx.x >> 5;                                          \
    const int lane = threadIdx.x & 31;                                          \
    const int half = lane >> 4;                                                 \
    const int lm   = lane & 15;                                                 \
    GEMM_STAGE_PROLOGUE(X, W, mBlock, n0, nrows_of_W);                          \
    for (int c = 0; c < KCHUNKS; ++c) {                                         \
        GEMM_STAGE_CHUNK(X, W, mBlock, n0, nrows_of_W, c);                      \
        const float* As = &ldsA[c & 1][(wave * 16 + lm) * LDSTR + 2 * half];    \
        const float* Bs = &ldsB[c & 1][lm * LDSTR + 2 * half];                  \
        _Pragma("unroll")                                                       \
        for (int k = 0; k < KC; k += 4) {                                       \
            v2f a;                                                              \
            a[0] = As[k];                                                       \
            a[1] = As[k + 1];                                                   \
            _Pragma("unroll")                                                   \
            for (int nt = 0; nt < 4; ++nt) {                                    \
                v2f b;                                                          \
                b[0] = Bs[nt * 16 * LDSTR + k];                                 \
                b[1] = Bs[nt * 16 * LDSTR + k + 1];                             \
                acc[nt] = __builtin_amdgcn_wmma_f32_16x16x4_f32(                \
                    false, a, false, b, (short)0, acc[nt], false, false);       \
            }                                                                   \
        }                                                                       \
    }

#ifdef USE_TDM
#define GEMM_STAGE_PROLOGUE(X, W, mBlock, n0, nrowsW)                           \
    if (wave == 0) {                                                            \
        tdm_load_tile(lds_off_of(&ldsA[0][0]), (X) + (size_t)(mBlock)*HID,      \
                      128, MTOT);                                               \
        tdm_load_tile(lds_off_of(&ldsB[0][0]), (W) + (size_t)(n0)*HID,          \
                      64, (nrowsW));                                            \
    }
#define GEMM_STAGE_CHUNK(X, W, mBlock, n0, nrowsW, c)                           \
    if (wave == 0) __builtin_amdgcn_s_wait_tensorcnt(0);                        \
    __syncthreads();                                                            \
    if (wave == 0 && (c) + 1 < KCHUNKS) {                                       \
        const int kn = ((c) + 1) * KC;                                          \
        tdm_load_tile(lds_off_of(&ldsA[((c) + 1) & 1][0]),                      \
                      (X) + (size_t)(mBlock)*HID + kn, 128, MTOT);              \
        tdm_load_tile(lds_off_of(&ldsB[((c) + 1) & 1][0]),                      \
                      (W) + (size_t)(n0)*HID + kn, 64, (nrowsW));               \
    }
#else
// Fallback: cooperative synchronous staging (single logical buffer per chunk).
#define GEMM_STAGE_PROLOGUE(X, W, mBlock, n0, nrowsW)
#define GEMM_STAGE_CHUNK(X, W, mBlock, n0, nrowsW, c)                           \
    {                                                                           \
        __syncthreads();                                                        \
        const int k0 = (c) * KC;                                                \
        const int g = threadIdx.x & 3, r = threadIdx.x >> 2;                    \
        _Pragma("unroll")                                                       \
        for (int it = 0; it < 2; ++it) {                                        \
            const int row = r + it * 64;                                        \
            *(float4*)&ldsA[(c) & 1][row * LDSTR + g * 4] =                     \
                *(const float4*)((X) + (size_t)((mBlock) + row) * HID + k0 +    \
                                 g * 4);                                        \
        }                                                                       \
        *(float4*)&ldsB[(c) & 1][r * LDSTR + g * 4] =                           \
            *(const float4*)((W) + (size_t)((n0) + r) * HID + k0 + g * 4);      \
        __syncthreads();                                                        \
    }
#endif

// -------------------- QKV GEMM (+bias, scatter to Q/K/V bufs) --------------------
__global__ __launch_bounds__(256) void qkv_gemm_kernel(
    const float* __restrict__ X, const float* __restrict__ W,
    const float* __restrict__ bias,
    float* __restrict__ Qb, float* __restrict__ Kb, float* __restrict__ Vb)
{
    const int mBlock = blockIdx.y * 128;
    const int n0 = blockIdx.x * 64;
    v8f acc[4] = {};
    GEMM_MAINLOOP(X, W, mBlock, n0, NQKV, acc)

    const int m0 = mBlock + wave * 16;
#pragma unroll
    for (int nt = 0; nt < 4; ++nt) {
#pragma unroll
        for (int r = 0; r < 8; ++r) {
            const int m = m0 + r + 8 * half;
            const int n = n0 + nt * 16 + lm;
            const float v = acc[nt][r] + bias[n];
            const int s = m / BATCH, b = m % BATCH;
            const int head = n / (3 * HD);
            const int rem  = n % (3 * HD);
            const int comp = rem / HD;
            const int d    = rem % HD;
            float* dst = (comp == 0) ? Qb : ((comp == 1) ? Kb : Vb);
            dst[(((size_t)b * HEADS + head) * SEQ + s) * HD + d] = v;
        }
    }
}

// -------------------- Dense GEMM (skip_bias_add) --------------------
__global__ __launch_bounds__(256) void dense_gemm_kernel(
    const float* __restrict__ X, const float* __restrict__ W,
    float* __restrict__ Out)
{
    const int mBlock = blockIdx.y * 128;
    const int n0 = blockIdx.x * 64;
    v8f acc[4] = {};
    GEMM_MAINLOOP(X, W, mBlock, n0, HID, acc)

    const int m0 = mBlock + wave * 16;
#pragma unroll
    for (int nt = 0; nt < 4; ++nt) {
#pragma unroll
        for (int r = 0; r < 8; ++r) {
            const int m = m0 + r + 8 * half;
            const int n = n0 + nt * 16 + lm;
            Out[(size_t)m * HID + n] = acc[nt][r];
        }
    }
}

// -------------------- RoPE (in place on Q and K bufs) --------------------
__global__ __launch_bounds__(256) void rope_kernel(float* __restrict__ Qb,
                                                   float* __restrict__ Kb)
{
    const int idx = blockIdx.x * blockDim.x + threadIdx.x;  // over B*H*S*64 pairs
    const int i  = idx & 63;
    const int s  = (idx >> 6) & (SEQ - 1);
    const int bh = idx >> 17;
    const float inv_freq = __powf(10000.0f, -(float)(2 * i) / 128.0f);
    const float ang = (float)s * inv_freq;
    const float c = __cosf(ang), sn = __sinf(ang);
    float* base = (blockIdx.y ? Kb : Qb) + ((size_t)bh * SEQ + s) * HD;
    const float x1 = base[i], x2 = base[i + 64];
    base[i]      = x1 * c - x2 * sn;
    base[i + 64] = x2 * c + x1 * sn;
}

// -------------------- Causal attention (flash style, fp32 WMMA) --------------------
__global__ __launch_bounds__(128) void attn_kernel(
    const float* __restrict__ Qb, const float* __restrict__ Kb,
    const float* __restrict__ Vb, float* __restrict__ Ctx)
{
    __shared__ float pl[4][16][17];  // per-wave P tile, padded
    const int wave = threadIdx.x >> 5;
    const int lane = threadIdx.x & 31;
    const int half = lane >> 4;
    const int lm   = lane & 15;
    const int bh = blockIdx.y;
    const int b = bh >> 4, h = bh & 15;
    const int q0 = blockIdx.x * 64 + wave * 16;

    const float* Q = Qb + (size_t)bh * SEQ * HD;
    const float* K = Kb + (size_t)bh * SEQ * HD;
    const float* V = Vb + (size_t)bh * SEQ * HD;

    // Q rows resident in registers in A-frag layout
    float qreg[64];
    {
        const float* qrow = Q + (size_t)(q0 + lm) * HD + 2 * half;
#pragma unroll
        for (int kc = 0; kc < 32; ++kc) {
            qreg[2 * kc]     = qrow[4 * kc];
            qreg[2 * kc + 1] = qrow[4 * kc + 1];
        }
    }

    v8f oacc[8] = {};
    float rowM[8], rowL[8];
#pragma unroll
    for (int r = 0; r < 8; ++r) { rowM[r] = -1e30f; rowL[r] = 0.0f; }

    const float scale = 0.08838834764831845f;  // 1/sqrt(128)

    for (int t0 = 0; t0 <= q0; t0 += 16) {
        // S tile = Q(16x128) x K^T(128x16)
        v8f sacc = {};
        const float* krow = K + (size_t)(t0 + lm) * HD + 2 * half;
#pragma unroll 8
        for (int kc = 0; kc < 32; ++kc) {
            v2f a;  a[0] = qreg[2 * kc];  a[1] = qreg[2 * kc + 1];
            v2f bb; bb[0] = krow[4 * kc]; bb[1] = krow[4 * kc + 1];
            sacc = __builtin_amdgcn_wmma_f32_16x16x4_f32(
                false, a, false, bb, (short)0, sacc, false, false);
        }
        // scale + causal mask (reference semantics: masked -> -10000 pre-softmax)
#pragma unroll
        for (int r = 0; r < 8; ++r) {
            const int sq = q0 + r + 8 * half;
            const int t  = t0 + lm;
            float sc = sacc[r] * scale;
            if (t > sq) sc = -10000.0f;
            sacc[r] = sc;
        }
        // online softmax per row (row r+8*half lives in this lane's half)
#pragma unroll
        for (int r = 0; r < 8; ++r) {
            float mx = sacc[r];
#pragma unroll
            for (int off = 1; off < 16; off <<= 1)
                mx = fmaxf(mx, __shfl_xor(mx, off, 32));
            const float Mnew = fmaxf(rowM[r], mx);
            const float alpha = __expf(rowM[r] - Mnew);
            rowL[r] *= alpha;
#pragma unroll
            for (int c = 0; c < 8; ++c) oacc[c][r] *= alpha;
            const float p = __expf(sacc[r] - Mnew);
            float ps = p;
#pragma unroll
            for (int off = 1; off < 16; off <<= 1)
                ps += __shfl_xor(ps, off, 32);
            rowL[r] += ps;
            rowM[r] = Mnew;
            pl[wave][r + 8 * half][lm] = p;  // C-layout store
        }
        asm volatile("s_wait_dscnt 0" ::: "memory");
        // P back in A-frag layout
        float pA[8];
#pragma unroll
        for (int ks = 0; ks < 4; ++ks) {
            pA[2 * ks]     = pl[wave][lm][4 * ks + 2 * half];
            pA[2 * ks + 1] = pl[wave][lm][4 * ks + 2 * half + 1];
        }
        // O += P(16x16) x V(16x128), 8 column chunks
#pragma unroll
        for (int c = 0; c < 8; ++c) {
            const float* vcol = V + (size_t)t0 * HD + c * 16 + lm;
#pragma unroll
            for (int ks = 0; ks < 4; ++ks) {
                v2f a;  a[0] = pA[2 * ks]; a[1] = pA[2 * ks + 1];
                const int kk = 4 * ks + 2 * half;
                v2f bb;
                bb[0] = vcol[(size_t)kk * HD];
                bb[1] = vcol[(size_t)(kk + 1) * HD];
                oacc[c] = __builtin_amdgcn_wmma_f32_16x16x4_f32(
                    false, a, false, bb, (short)0, oacc[c], false, false);
            }
        }
    }
    // normalize and write context in [s, b, h*128+d] layout for the dense GEMM
#pragma unroll
    for (int c = 0; c < 8; ++c) {
#pragma unroll
        for (int r = 0; r < 8; ++r) {
            const int s = q0 + r + 8 * half;
            const int d = c * 16 + lm;
            Ctx[((size_t)s * BATCH + b) * HID + h * HD + d] = oacc[c][r] / rowL[r];
        }
    }
}

__global__ void bias_copy_kernel(const float* __restrict__ b, float* __restrict__ dst)
{
    const int i = blockIdx.x * blockDim.x + threadIdx.x;
    if (i < HID) dst[i] = b[i];
}

extern "C" void kernel_launch(void* const* d_in, const int* in_sizes, int n_in,
                              void* d_out, int out_size, void* d_ws, size_t ws_size,
                              hipStream_t stream) {
    const float* hidden = (const float*)d_in[0];
    // d_in[1] = attention_mask (causal; computed analytically, unused)
    const float* w_qkv   = (const float*)d_in[2];
    const float* b_qkv   = (const float*)d_in[3];
    const float* w_dense = (const float*)d_in[4];
    const float* b_dense = (const float*)d_in[5];
    float* out = (float*)d_out;

    float* Qb  = (float*)d_ws;
    float* Kb  = Qb + BHSD;
    float* Vb  = Kb + BHSD;
    float* Ctx = Vb + BHSD;

    // QKV GEMM: M=4096, N=6144 -> grid (6144/64, 4096/128)
    qkv_gemm_kernel<<<dim3(NQKV / 64, MTOT / 128), dim3(256), 0, stream>>>(
        hidden, w_qkv, b_qkv, Qb, Kb, Vb);

    // RoPE: B*H*S*64 pairs = 4194304 threads, y-dim selects Q vs K
    rope_kernel<<<dim3((BATCH * HEADS * SEQ * 64) / 256, 2), dim3(256), 0, stream>>>(Qb, Kb);

    // Attention: (query blocks of 64, b*h)
    attn_kernel<<<dim3(SEQ / 64, BATCH * HEADS), dim3(128), 0, stream>>>(Qb, Kb, Vb, Ctx);

    // Dense GEMM: M=4096, N=2048
    dense_gemm_kernel<<<dim3(HID / 64, MTOT / 128), dim3(256), 0, stream>>>(Ctx, w_dense, out);

    // b_dense appended after the main output
    bias_copy_kernel<<<HID / 256, 256, 0, stream>>>(b_dense, out + (size_t)MTOT * HID);
}